// GCN_29454885716255
// MI455X (gfx1250) — compile-verified
//
#include <hip/hip_runtime.h>
#include <math.h>

typedef __attribute__((ext_vector_type(2))) float v2f;
typedef __attribute__((ext_vector_type(8))) float v8f;

// ---------------- utility kernels ----------------

__global__ void gcn_fill_kernel(float* __restrict__ p, float v, long long count) {
    long long i = (long long)blockIdx.x * blockDim.x + threadIdx.x;
    if (i < count) p[i] = v;
}

// deg[dst] += 1 per edge (deg pre-initialized to 1.0 for the self loop)
__global__ void gcn_degree_kernel(const long long* __restrict__ dst, long long E,
                                  float* __restrict__ deg) {
    long long e = (long long)blockIdx.x * blockDim.x + threadIdx.x;
    if (e < E) atomicAdd(&deg[dst[e]], 1.0f);
}

// dis = deg^-0.5 in place (deg >= 1 always due to self loops)
__global__ void gcn_rsqrt_kernel(float* __restrict__ deg, long long n) {
    long long i = (long long)blockIdx.x * blockDim.x + threadIdx.x;
    if (i < n) deg[i] = 1.0f / sqrtf(deg[i]);
}

// ---------------- WMMA node transform: m = h @ W ----------------
// One wave32 handles a 16-node tile. f32 WMMA 16x16x4, chained over K in
// chunks of 4 (fully unrolled via template). A: lane = row (M), VGPR0/1 hold
// K = {2*hi, 2*hi+1} of the chunk. B: lane = column (N), same K pair;
// columns >= D_OUT are zeroed via select (loads use a clamped valid address,
// so no EXEC-mask predication around the loads). D: VGPR j -> row j + 8*hi,
// lane%16 -> column.
template <int D_IN, int D_OUT>
__global__ void gcn_transform_wmma_kernel(const float* __restrict__ h,
                                          const float* __restrict__ W,  // [D_IN, D_OUT] row-major
                                          float* __restrict__ m,        // [n, D_OUT]
                                          int n) {
    const int lane  = threadIdx.x & 31;
    const int wave  = threadIdx.x >> 5;
    const int tile  = blockIdx.x * (blockDim.x >> 5) + wave;
    const int node0 = tile * 16;
    if (node0 >= n) return;               // wave-uniform exit: EXEC stays all-1s

    const int row = lane & 15;
    const int hi  = lane >> 4;
    const int col = row;                        // this lane's B/D column
    const int wcol = (col < D_OUT) ? col : (D_OUT - 1);  // clamped valid address
    const bool colok = (col < D_OUT);

    int arow = node0 + row;
    if (arow >= n) arow = n - 1;          // clamp: safe load, store is guarded

    const float* __restrict__ hr = h + (long long)arow * D_IN;

    v8f acc = {};
#pragma unroll
    for (int c = 0; c < D_IN; c += 4) {
        const int k0 = c + 2 * hi;
        v2f a, b;
        a.x = hr[k0];
        a.y = hr[k0 + 1];
        float w0 = W[k0 * D_OUT + wcol];        // unconditional load
        float w1 = W[(k0 + 1) * D_OUT + wcol];  // unconditional load
        b.x = colok ? w0 : 0.0f;                // v_cndmask, no EXEC churn
        b.y = colok ? w1 : 0.0f;
        acc = __builtin_amdgcn_wmma_f32_16x16x4_f32(
            /*neg_a=*/false, a, /*neg_b=*/false, b,
            /*c_mod=*/(short)0, acc, /*reuse_a=*/false, /*reuse_b=*/false);
    }

    if (colok) {
#pragma unroll
        for (int j = 0; j < 8; ++j) {
            int node = node0 + j + 8 * hi;
            if (node < n) m[(long long)node * D_OUT + col] = acc[j];
        }
    }
}

// ---------------- edge scatter: agg[dst] += dis[s]*dis[d] * m[src] ----------------
// Templated on D_OUT: message read is one b128/b64 load, atomics fully unrolled.
template <int D_OUT>
__global__ void gcn_scatter_kernel(const long long* __restrict__ src,
                                   const long long* __restrict__ dst,
                                   const float* __restrict__ dis,
                                   const float* __restrict__ m,
                                   float* __restrict__ agg,
                                   long long E) {
    long long e = (long long)blockIdx.x * blockDim.x + threadIdx.x;
    if (e >= E) return;
    const long long s = src[e];
    const long long d = dst[e];
    const float w = dis[s] * dis[d];
    float* ad = agg + d * D_OUT;
    if (D_OUT == 4) {
        const float4 mv = *(const float4*)(m + s * 4);   // global_load_b128
        atomicAdd(&ad[0], w * mv.x);
        atomicAdd(&ad[1], w * mv.y);
        atomicAdd(&ad[2], w * mv.z);
        atomicAdd(&ad[3], w * mv.w);
    } else {
        const float2 mv = *(const float2*)(m + s * 2);   // global_load_b64
        atomicAdd(&ad[0], w * mv.x);
        atomicAdd(&ad[1], w * mv.y);
    }
}

// ---------------- finalize: h = tanh(agg + dis^2 * m + b) (self-loop folded in) ----
// One thread per node; vectorized b128/b64 access.
template <int D_OUT>
__global__ void gcn_finalize_kernel(float* __restrict__ agg,
                                    const float* __restrict__ m,
                                    const float* __restrict__ dis,
                                    const float* __restrict__ b,
                                    long long n) {
    long long i = (long long)blockIdx.x * blockDim.x + threadIdx.x;
    if (i >= n) return;
    const float w2 = dis[i] * dis[i];
    if (D_OUT == 4) {
        float4 a = *(float4*)(agg + i * 4);
        const float4 mv = *(const float4*)(m + i * 4);
        a.x = tanhf(a.x + w2 * mv.x + b[0]);
        a.y = tanhf(a.y + w2 * mv.y + b[1]);
        a.z = tanhf(a.z + w2 * mv.z + b[2]);
        a.w = tanhf(a.w + w2 * mv.w + b[3]);
        *(float4*)(agg + i * 4) = a;
    } else {
        float2 a = *(float2*)(agg + i * 2);
        const float2 mv = *(const float2*)(m + i * 2);
        a.x = tanhf(a.x + w2 * mv.x + b[0]);
        a.y = tanhf(a.y + w2 * mv.y + b[1]);
        *(float2*)(agg + i * 2) = a;
    }
}

// ---------------- global mean pool accumulation ----------------
__global__ void gcn_pool_kernel(const float* __restrict__ h,   // [n, 2]
                                const long long* __restrict__ batch,
                                float* __restrict__ sums,      // [G, 2]
                                float* __restrict__ cnt,       // [G]
                                long long n) {
    long long i = (long long)blockIdx.x * blockDim.x + threadIdx.x;
    if (i >= n) return;
    long long g = batch[i];
    const float2 hv = *(const float2*)(h + i * 2);
    atomicAdd(&sums[g * 2 + 0], hv.x);
    atomicAdd(&sums[g * 2 + 1], hv.y);
    atomicAdd(&cnt[g], 1.0f);
}

// ---------------- head: hg = sums/max(cnt,1); out = hg @ Wc + bc ----------------
// d_out layout: [0, G) = out, [G, 3G) = hg row-major
__global__ void gcn_head_kernel(const float* __restrict__ sums,
                                const float* __restrict__ cnt,
                                const float* __restrict__ Wc,
                                const float* __restrict__ bc,
                                float* __restrict__ out, int G) {
    int g = blockIdx.x * blockDim.x + threadIdx.x;
    if (g >= G) return;
    float c = fmaxf(cnt[g], 1.0f);
    float h0 = sums[g * 2 + 0] / c;
    float h1 = sums[g * 2 + 1] / c;
    out[g] = h0 * Wc[0] + h1 * Wc[1] + bc[0];
    out[G + g * 2 + 0] = h0;
    out[G + g * 2 + 1] = h1;
}

// ---------------- launch ----------------

static inline int cdiv_ll(long long a, long long b) { return (int)((a + b - 1) / b); }

extern "C" void kernel_launch(void* const* d_in, const int* in_sizes, int n_in,
                              void* d_out, int out_size, void* d_ws, size_t ws_size,
                              hipStream_t stream) {
    const float*     x     = (const float*)d_in[0];
    const long long* eidx  = (const long long*)d_in[1];
    const long long* batch = (const long long*)d_in[2];
    const float* W1 = (const float*)d_in[3];
    const float* b1 = (const float*)d_in[4];
    const float* W2 = (const float*)d_in[5];
    const float* b2 = (const float*)d_in[6];
    const float* W3 = (const float*)d_in[7];
    const float* b3 = (const float*)d_in[8];
    const float* Wc = (const float*)d_in[9];
    const float* bc = (const float*)d_in[10];

    const long long n = in_sizes[0] / 16;   // F = 16
    const long long E = in_sizes[1] / 2;
    const int G = out_size / 3;             // out(G) + hg(2G)

    const long long* src = eidx;            // edge_index[0]
    const long long* dst = eidx + E;        // edge_index[1]

    // workspace layout
    float* P    = (float*)d_ws;             // n*4  (messages m)
    float* Q    = P + n * 4;                // n*4  (agg / hidden, reused)
    float* dis  = Q + n * 4;                // n    (deg -> deg^-1/2)
    float* sums = dis + n;                  // G*2
    float* cnt  = sums + (long long)G * 2;  // G

    const int TB = 256;
    float* outp = (float*)d_out;

    // degree + normalization
    gcn_fill_kernel<<<cdiv_ll(n, TB), TB, 0, stream>>>(dis, 1.0f, n);
    gcn_degree_kernel<<<cdiv_ll(E, TB), TB, 0, stream>>>(dst, E, dis);
    gcn_rsqrt_kernel<<<cdiv_ll(n, TB), TB, 0, stream>>>(dis, n);

    const int tiles = cdiv_ll(n, 16);
    const int tgrid = (tiles + 7) / 8;      // 8 waves (tiles) per 256-thread block

    // layer 1: x[n,16] -> Q[n,4]
    gcn_transform_wmma_kernel<16, 4><<<tgrid, 256, 0, stream>>>(x, W1, P, (int)n);
    gcn_fill_kernel<<<cdiv_ll(n * 4, TB), TB, 0, stream>>>(Q, 0.0f, n * 4);
    gcn_scatter_kernel<4><<<cdiv_ll(E, TB), TB, 0, stream>>>(src, dst, dis, P, Q, E);
    gcn_finalize_kernel<4><<<cdiv_ll(n, TB), TB, 0, stream>>>(Q, P, dis, b1, n);

    // layer 2: Q[n,4] -> Q[n,4]
    gcn_transform_wmma_kernel<4, 4><<<tgrid, 256, 0, stream>>>(Q, W2, P, (int)n);
    gcn_fill_kernel<<<cdiv_ll(n * 4, TB), TB, 0, stream>>>(Q, 0.0f, n * 4);
    gcn_scatter_kernel<4><<<cdiv_ll(E, TB), TB, 0, stream>>>(src, dst, dis, P, Q, E);
    gcn_finalize_kernel<4><<<cdiv_ll(n, TB), TB, 0, stream>>>(Q, P, dis, b2, n);

    // layer 3: Q[n,4] -> Q[n,2]
    gcn_transform_wmma_kernel<4, 2><<<tgrid, 256, 0, stream>>>(Q, W3, P, (int)n);
    gcn_fill_kernel<<<cdiv_ll(n * 2, TB), TB, 0, stream>>>(Q, 0.0f, n * 2);
    gcn_scatter_kernel<2><<<cdiv_ll(E, TB), TB, 0, stream>>>(src, dst, dis, P, Q, E);
    gcn_finalize_kernel<2><<<cdiv_ll(n, TB), TB, 0, stream>>>(Q, P, dis, b3, n);

    // mean pool + head
    gcn_fill_kernel<<<cdiv_ll((long long)G * 2, TB), TB, 0, stream>>>(sums, 0.0f, (long long)G * 2);
    gcn_fill_kernel<<<cdiv_ll(G, TB), TB, 0, stream>>>(cnt, 0.0f, G);
    gcn_pool_kernel<<<cdiv_ll(n, TB), TB, 0, stream>>>(Q, batch, sums, cnt, n);
    gcn_head_kernel<<<cdiv_ll(G, TB), TB, 0, stream>>>(sums, cnt, Wc, bc, outp, G);
}